// MultiLossFactory_42812234006947
// MI455X (gfx1250) — compile-verified
//
#include <hip/hip_runtime.h>
#include <hip/hip_bf16.h>
#include <math.h>

// Problem geometry (fixed by the reference): pred/gt are [2, 8, 64, 128, 128] f32.
#define DIM_D 64
#define DIM_H 128
#define DIM_W 128
#define NSLICE 16                   // B*C
#define SLICE  (DIM_D*DIM_H*DIM_W)  // 1,048,576 elements per (b,c) slice
#define SLICE4 (SLICE/4)            // 262,144 float4s per slice
#define BLK_A  128                  // blocks per slice, pass A
#define BLK_B  128                  // blocks per (tensor,slice), pass B
#define TPB    256                  // 8 wave32s per block
#define ITERS_A (SLICE4 / BLK_A / TPB)   // 8, compile-time
#define ITERS_B (SLICE4 / BLK_B / TPB)   // 8, compile-time

#define SMOOTH_L1_BETA (1.0f/9.0f)

// Workspace layout (floats / uint32 keys):
//  [0]                      heatmap weighted-Huber global sum (atomicAdd)
//  [1..16]                  ordered-uint max keys, pred slices (atomicMax)
//  [17..32]                 ordered-uint max keys, gt   slices (atomicMax)
//  [33 + (t*16+s)*4 + k]    k in {S, Sx, Sy, Sz} exp-weighted sums (atomicAdd)
#define WS_HSUM 0
#define WS_MAXP 1
#define WS_MAXG 17
#define WS_SUMS 33
#define WS_TOTAL (33 + 32*4)

typedef float v2f __attribute__((ext_vector_type(2)));
typedef float v8f __attribute__((ext_vector_type(8)));

// ---- float <-> monotone uint key (for atomicMax on floats of any sign) ----
__device__ __forceinline__ unsigned fkey(float f) {
    unsigned u = __float_as_uint(f);
    return (u & 0x80000000u) ? ~u : (u | 0x80000000u);
}
__device__ __forceinline__ float funkey(unsigned k) {
    unsigned u = (k & 0x80000000u) ? (k & 0x7FFFFFFFu) : ~k;
    return __uint_as_float(u);
}

// ---- wave32 sum reduction; first stage is a real v_wmma_f32_16x16x4_f32 ----
// A = ones(16x4), B row-striped with lane values in VGPR b[0] and zeros in b[1]
// => D[m,n] = v[n] + v[n+16] for every m; c[0] holds that column sum in all
// lane groups, so 4 butterfly steps finish the remaining 16-wide reduction.
// Caller must have EXEC all-ones (uniform control flow) per WMMA rules.
__device__ __forceinline__ float wave_reduce_sum(float v) {
    v2f a; a[0] = 1.0f; a[1] = 1.0f;
    v2f b; b[0] = v;    b[1] = 0.0f;
    v8f c = {};
    c = __builtin_amdgcn_wmma_f32_16x16x4_f32(false, a, false, b, (short)0, c,
                                              false, false);
    float s = c[0];
    s += __shfl_xor(s, 1, 32);
    s += __shfl_xor(s, 2, 32);
    s += __shfl_xor(s, 4, 32);
    s += __shfl_xor(s, 8, 32);
    return s;
}

__device__ __forceinline__ float wave_reduce_max(float v) {
    v = fmaxf(v, __shfl_xor(v, 16, 32));
    v = fmaxf(v, __shfl_xor(v, 8, 32));
    v = fmaxf(v, __shfl_xor(v, 4, 32));
    v = fmaxf(v, __shfl_xor(v, 2, 32));
    v = fmaxf(v, __shfl_xor(v, 1, 32));
    return v;
}

// Block sum: WMMA wave reduce -> LDS -> thread 0 combines the 8 wave partials.
// Result valid in thread 0 only.
__device__ __forceinline__ float block_sum(float v, float* sh) {
    v = wave_reduce_sum(v);
    const int wid = threadIdx.x >> 5, lane = threadIdx.x & 31;
    if (lane == 0) sh[wid] = v;
    __syncthreads();
    float r = 0.0f;
    if (threadIdx.x == 0) {
        #pragma unroll
        for (int i = 0; i < TPB / 32; ++i) r += sh[i];
    }
    __syncthreads();   // allow LDS buffer reuse by the next reduction
    return r;
}

__device__ __forceinline__ float block_max(float v, float* sh) {
    v = wave_reduce_max(v);
    const int wid = threadIdx.x >> 5, lane = threadIdx.x & 31;
    if (lane == 0) sh[wid] = v;
    __syncthreads();
    float r = -__builtin_huge_valf();
    if (threadIdx.x == 0) {
        #pragma unroll
        for (int i = 0; i < TPB / 32; ++i) r = fmaxf(r, sh[i]);
    }
    __syncthreads();
    return r;
}

__device__ __forceinline__ float huber_weighted(float p, float g) {
    float err  = fabsf(p - g);
    float quad = fminf(err, 1.0f);          // clip(err, 0, delta=1)
    float lin  = err - quad;
    float basic = 0.5f * quad * quad + lin; // delta == 1
    float w = (g != 0.0f) ? 1.5f : 0.6f;
    return w * basic;
}

// ---------------- init: zero the accumulator workspace every launch --------
__global__ void init_ws(float* ws) {
    int i = blockIdx.x * blockDim.x + threadIdx.x;
    if (i < WS_TOTAL) ws[i] = 0.0f;   // key 0 < key(any finite float)
}

// ---------------- pass A: fused Huber partials + per-slice maxes -----------
// Reads 128 MiB (pred + gt) once; this is the HBM-bound pass (~5.5us ideal).
__global__ void pass_a(const float* __restrict__ pred,
                       const float* __restrict__ gt, float* ws) {
    __shared__ float sh[TPB / 32];
    const int s   = blockIdx.x / BLK_A;
    const int blk = blockIdx.x % BLK_A;
    const int per_block = SLICE4 / BLK_A;          // 2048 float4s
    const float4* p4 = (const float4*)pred + (size_t)s * SLICE4 + (size_t)blk * per_block;
    const float4* g4 = (const float4*)gt   + (size_t)s * SLICE4 + (size_t)blk * per_block;

    float hacc = 0.0f;
    float mp = -__builtin_huge_valf(), mg = -__builtin_huge_valf();

    #pragma unroll
    for (int k = 0; k < ITERS_A; ++k) {            // compile-time trip count
        const int i = threadIdx.x + k * TPB;
        if (k + 1 < ITERS_A) {                     // folds away at compile time
            // WGP-scope prefetch of the next chunk (in-bounds by the guard)
            __builtin_prefetch((const void*)(p4 + i + TPB), 0, 3);
            __builtin_prefetch((const void*)(g4 + i + TPB), 0, 3);
        }
        float4 p = p4[i];
        float4 g = g4[i];
        hacc += huber_weighted(p.x, g.x) + huber_weighted(p.y, g.y)
              + huber_weighted(p.z, g.z) + huber_weighted(p.w, g.w);
        mp = fmaxf(mp, fmaxf(fmaxf(p.x, p.y), fmaxf(p.z, p.w)));
        mg = fmaxf(mg, fmaxf(fmaxf(g.x, g.y), fmaxf(g.z, g.w)));
    }

    float hb  = block_sum(hacc, sh);
    float mpb = block_max(mp, sh);
    float mgb = block_max(mg, sh);
    if (threadIdx.x == 0) {
        atomicAdd(&ws[WS_HSUM], hb);
        atomicMax((unsigned*)ws + WS_MAXP + s, fkey(mpb));
        atomicMax((unsigned*)ws + WS_MAXG + s, fkey(mgb));
    }
}

// ---------------- pass B: softmax exp sums + coordinate moments ------------
// Re-reads 128 MiB; both tensors fit in the 192 MB L2 so this pass is
// L2-resident, not HBM-bound.
__global__ void pass_b(const float* __restrict__ pred,
                       const float* __restrict__ gt, float* ws) {
    __shared__ float sh[TPB / 32];
    const int pair = blockIdx.x / BLK_B;   // 0..31  (t*16 + s)
    const int blk  = blockIdx.x % BLK_B;
    const int t = pair >> 4, s = pair & 15;
    const float* base = (t == 0 ? pred : gt) + (size_t)s * SLICE;
    const float mx = funkey(((const unsigned*)ws)[(t == 0 ? WS_MAXP : WS_MAXG) + s]);

    const int per_block = SLICE4 / BLK_B;  // 2048 float4s
    const float4* x4 = (const float4*)base + (size_t)blk * per_block;
    const int base4 = blk * per_block;

    float S = 0.0f, Sx = 0.0f, Sy = 0.0f, Sz = 0.0f;
    #pragma unroll
    for (int k = 0; k < ITERS_B; ++k) {            // compile-time trip count
        const int i = threadIdx.x + k * TPB;
        if (k + 1 < ITERS_B)
            __builtin_prefetch((const void*)(x4 + i + TPB), 0, 3);
        float4 x = x4[i];
        const int idx = (base4 + i) << 2;          // element index in slice
        const int d = idx >> 14;                   // / (H*W)
        const int h = (idx >> 7) & 127;
        const int w = idx & 127;                   // W divisible by 4: no wrap
        float e0 = __expf(x.x - mx);
        float e1 = __expf(x.y - mx);
        float e2 = __expf(x.z - mx);
        float e3 = __expf(x.w - mx);
        float es = (e0 + e1) + (e2 + e3);
        S  += es;
        Sx += es * ((float)(d + 1) * (1.0f / (float)DIM_D));
        Sy += es * ((float)(h + 1) * (1.0f / (float)DIM_H));
        Sz += (e0 * (float)(w + 1) + e1 * (float)(w + 2) +
               e2 * (float)(w + 3) + e3 * (float)(w + 4)) * (1.0f / (float)DIM_W);
    }

    float Sb  = block_sum(S,  sh);
    float Sxb = block_sum(Sx, sh);
    float Syb = block_sum(Sy, sh);
    float Szb = block_sum(Sz, sh);
    if (threadIdx.x == 0) {
        float* dst = ws + WS_SUMS + pair * 4;
        atomicAdd(dst + 0, Sb);
        atomicAdd(dst + 1, Sxb);
        atomicAdd(dst + 2, Syb);
        atomicAdd(dst + 3, Szb);
    }
}

// ---------------- pass C: finalize scalar ---------------------------------
__global__ void pass_c(const float* __restrict__ ws, float* __restrict__ out) {
    if (threadIdx.x != 0 || blockIdx.x != 0) return;
    float pts[2][NSLICE][3];
    for (int t = 0; t < 2; ++t) {
        for (int s = 0; s < NSLICE; ++s) {
            const float* src = ws + WS_SUMS + (t * NSLICE + s) * 4;
            float S = src[0];
            pts[t][s][0] = (src[1] / S) * (float)DIM_D - 1.0f;
            pts[t][s][1] = (src[2] / S) * (float)DIM_H - 1.0f;
            pts[t][s][2] = (src[3] / S) * (float)DIM_W - 1.0f;
        }
    }
    float off = 0.0f;
    for (int s = 0; s < NSLICE; ++s) {
        for (int k = 0; k < 3; ++k) {
            float l1 = fabsf(pts[0][s][k] - pts[1][s][k]);
            off += (l1 < SMOOTH_L1_BETA) ? 0.5f * l1 * l1 / SMOOTH_L1_BETA
                                         : l1 - 0.5f * SMOOTH_L1_BETA;
        }
    }
    off /= (float)NSLICE;                       // / (B*C)
    float heat = ws[WS_HSUM] / (float)NSLICE;   // mean over (b,c)
    float ratio = off / heat;                   // stop_gradient: value only
    out[0] = heat + off / ratio;
}

extern "C" void kernel_launch(void* const* d_in, const int* in_sizes, int n_in,
                              void* d_out, int out_size, void* d_ws, size_t ws_size,
                              hipStream_t stream) {
    const float* pred = (const float*)d_in[0];
    const float* gt   = (const float*)d_in[1];
    float* ws  = (float*)d_ws;
    float* out = (float*)d_out;
    (void)in_sizes; (void)n_in; (void)out_size; (void)ws_size;

    hipLaunchKernelGGL(init_ws, dim3(1), dim3(256), 0, stream, ws);
    hipLaunchKernelGGL(pass_a,  dim3(NSLICE * BLK_A), dim3(TPB), 0, stream, pred, gt, ws);
    hipLaunchKernelGGL(pass_b,  dim3(2 * NSLICE * BLK_B), dim3(TPB), 0, stream, pred, gt, ws);
    hipLaunchKernelGGL(pass_c,  dim3(1), dim3(32), 0, stream, ws, out);
}